// ActionDetokenizer_2345052143782
// MI455X (gfx1250) — compile-verified
//
#include <hip/hip_runtime.h>

// fp32 WMMA fragment types for V_WMMA_F32_16X16X4_F32:
// A (16x4 f32) and B (4x16 f32) are 2 VGPRs/lane, C/D (16x16 f32) is 8 VGPRs/lane.
typedef __attribute__((ext_vector_type(2))) float v2f;
typedef __attribute__((ext_vector_type(8))) float v8f;

#define NUM_JOINTS 12
#define NODES      13
#define EMBED      256
#define OUT_K      2
#define OUT_STRIDE (NUM_JOINTS * OUT_K)   // 24 floats per batch row
#define ROWSTRIDE  68                     // 64 + 4 pad: 272B rows -> 16B aligned, stride%64==4 -> conflict-free columns
#define WELEMS     (NUM_JOINTS * EMBED * OUT_K)  // 6144
#define WPAD       6160                   // covers j*512 + d*2 + n for n up to 15; tail zeroed

__global__ __launch_bounds__(256) void actdetok_wmma_f32(
    const float* __restrict__ x,     // (65536, 13, 256)
    const float* __restrict__ W,     // (12, 256, 2)
    const float* __restrict__ bias,  // (12, 2)
    float* __restrict__ out,         // (65536, 12, 2)
    int Bsz)
{
    __shared__ float ldsW[WPAD];                    // 24,640 B
    __shared__ float ldsX[8 * 16 * ROWSTRIDE];      // 34,816 B (per-wave 16x64 staging tiles)

    const int tid  = threadIdx.x;
    const int lane = tid & 31;
    const int wave = tid >> 5;
    const int hl   = lane >> 4;      // which 16-lane half of the wave32
    const int mn   = lane & 15;      // M index (A/C rows) or N index (B/C cols)

    // Stage all of W into LDS once per block; zero the pad tail so masked lanes
    // never multiply garbage.
    for (int i = tid; i < WPAD; i += 256)
        ldsW[i] = (i < WELEMS) ? W[i] : 0.0f;
    __syncthreads();

    const int b0 = blockIdx.x * 128 + wave * 16;    // this wave's 16-row batch tile
    if (b0 >= Bsz) return;
    float* ldsXw = &ldsX[wave * 16 * ROWSTRIDE];
    const float wmask = (mn < OUT_K) ? 1.0f : 0.0f; // zero B columns n>=2 (select via VALU, EXEC stays full)

    for (int j = 0; j < NUM_JOINTS; ++j) {
        const int node = j + 1;                      // JOINT_TO_NODE[j] == j+1
        v8f c = {};                                  // fp32 accumulator, 16x16 tile

        for (int kc = 0; kc < 4; ++kc) {             // K chunks of 64
            // ---- coalesced global -> LDS staging: 16 rows x 64 floats ----
            // 8 iterations x 32 lanes x float4 = 1024 floats. Lanes 0..15 cover
            // 256 contiguous bytes of one row -> clean global_load_b128 bursts.
            #pragma unroll
            for (int it = 0; it < 8; ++it) {
                const int idx = it * 32 + lane;
                const int row = idx >> 4;            // 0..15
                const int c4  = idx & 15;            // float4 slot within 64-float chunk
                const long gofs = (((long)(b0 + row)) * NODES + node) * EMBED + kc * 64 + c4 * 4;
                const float4 v = *(const float4*)(x + gofs);
                *(float4*)(ldsXw + row * ROWSTRIDE + c4 * 4) = v;
            }
            // Prefetch the next chunk's cachelines while we crunch this one
            // (lowers to global_prefetch_b8; speculative, no counter cost).
            if (kc < 3) {
                const long pofs = (((long)(b0 + (lane >> 1))) * NODES + node) * EMBED
                                  + (kc + 1) * 64 + (lane & 1) * 32;
                __builtin_prefetch(x + pofs, 0, 1);
            }

            // ---- 16 WMMA K-steps over this 64-wide chunk ----
            #pragma unroll
            for (int ks = 0; ks < 16; ++ks) {
                // A fragment: A[m, k] = x[b0+m, node, d0+k]
                //   lane (hl,mn): VGPR0 = A[mn, 2*hl], VGPR1 = A[mn, 2*hl+1]
                const int dl = ks * 4 + 2 * hl;      // K offset within chunk (byte addr 8-aligned)
                const v2f a = *(const v2f*)(ldsXw + mn * ROWSTRIDE + dl);

                // B fragment: B[k, n] = W[j, d0+k, n] (n<2), else 0
                //   lane (hl,mn): VGPR0 = B[2*hl, mn], VGPR1 = B[2*hl+1, mn]
                const int d0 = kc * 64 + dl;
                v2f bm;
                bm.x = ldsW[j * (EMBED * OUT_K) + d0 * OUT_K + mn] * wmask;
                bm.y = ldsW[j * (EMBED * OUT_K) + (d0 + 1) * OUT_K + mn] * wmask;

                // D = A x B + C  (16x16x4 fp32; exact-precision path, no downcast)
                c = __builtin_amdgcn_wmma_f32_16x16x4_f32(
                        /*neg_a=*/false, a, /*neg_b=*/false, bm,
                        /*c_mod=*/(short)0, c, /*reuse_a=*/false, /*reuse_b=*/false);
            }
        }

        // C layout: VGPR v holds M = v + 8*hl, N = mn. Only N<2 is real output.
        if (mn < OUT_K) {
            const float bb = bias[j * OUT_K + mn];
            #pragma unroll
            for (int v = 0; v < 8; ++v) {
                const int row = b0 + v + 8 * hl;
                out[(long)row * OUT_STRIDE + j * OUT_K + mn] = c[v] + bb;
            }
        }
    }
}

extern "C" void kernel_launch(void* const* d_in, const int* in_sizes, int n_in,
                              void* d_out, int out_size, void* d_ws, size_t ws_size,
                              hipStream_t stream) {
    const float* x  = (const float*)d_in[0];   // (B, 13, 256) f32
    const float* W  = (const float*)d_in[1];   // (12, 256, 2) f32
    const float* b  = (const float*)d_in[2];   // (12, 2) f32
    float* out      = (float*)d_out;           // (B, 12, 2) f32

    const int B = in_sizes[0] / (NODES * EMBED);   // 65536
    const int grid = (B + 127) / 128;              // 128 batch rows per 256-thread block
    actdetok_wmma_f32<<<grid, 256, 0, stream>>>(x, W, b, out, B);
}